// GNNLayer_64252710748237
// MI455X (gfx1250) — compile-verified
//
#include <hip/hip_runtime.h>
#include <hip/hip_bf16.h>

#define N_NODES 50000
#define N_EDGES 600000
#define DIM     128

typedef __attribute__((ext_vector_type(2))) float v2f;
typedef __attribute__((ext_vector_type(8))) float v8f;

// ---------------- degree / normalization ----------------

__global__ void k_init_deg(float* __restrict__ deg) {
    int i = blockIdx.x * blockDim.x + threadIdx.x;
    if (i < N_NODES) deg[i] = 1.0f;               // self-loop contributes 1
}

__global__ void k_count_deg(const int* __restrict__ ei, float* __restrict__ deg) {
    int e = blockIdx.x * blockDim.x + threadIdx.x;
    if (e < N_EDGES) {
        int d = ei[N_EDGES + e];                  // edge_index[1][e]
        unsafeAtomicAdd(&deg[d], 1.0f);           // hw global_atomic_add_f32
    }
}

__global__ void k_dinv(float* __restrict__ deg) {
    int i = blockIdx.x * blockDim.x + threadIdx.x;
    if (i < N_NODES) deg[i] = __frsqrt_rn(deg[i]); // deg >= 1 always
}

// ---------------- h = x @ W via V_WMMA_F32_16X16X4_F32 ----------------
// One wave computes one 16x16 tile of h. 8 waves/block = the 8 column tiles
// of one 16-row strip. Grid = 50000/16 = 3125 strips.

__global__ __launch_bounds__(256) void k_gemm(const float* __restrict__ x,
                                              const float* __restrict__ W,
                                              float* __restrict__ h) {
    const int wave = threadIdx.x >> 5;            // 0..7 -> column tile
    const int lane = threadIdx.x & 31;
    const int half = lane >> 4;                   // 0: K pair {0,1}, 1: {2,3}
    const int l    = lane & 15;                   // M (for A) / N (for B,C,D)
    const int row0 = blockIdx.x * 16;
    const int col0 = wave * 16;

    // A fragment source: x[row0+l][k0 + 2*half + {0,1}]  (8B aligned)
    const float* arow = x + (size_t)(row0 + l) * DIM + 2 * half;
    // B fragment source: W[k0 + 2*half + {0,1}][col0 + l]
    const float* bcol = W + (size_t)(2 * half) * DIM + col0 + l;

    v8f c = {0.f, 0.f, 0.f, 0.f, 0.f, 0.f, 0.f, 0.f};

#pragma unroll
    for (int k0 = 0; k0 < DIM; k0 += 4) {
        v2f a = *(const v2f*)(arow + k0);
        v2f b;
        b.x = bcol[(size_t)k0 * DIM];
        b.y = bcol[(size_t)(k0 + 1) * DIM];
        // 8 args: (neg_a, A, neg_b, B, c_mod, C, reuse_a, reuse_b)
        c = __builtin_amdgcn_wmma_f32_16x16x4_f32(false, a, false, b,
                                                  (short)0, c, false, false);
    }

    // C/D layout: VGPR v holds M = v + 8*half, N = l
    float* hp = h + (size_t)(row0 + 8 * half) * DIM + col0 + l;
#pragma unroll
    for (int v = 0; v < 8; ++v)
        hp[(size_t)v * DIM] = c[v];
}

// ---------------- out = h * dinv^2 + b  (self-loop + bias) ----------------

__global__ void k_init_out(const float* __restrict__ h,
                           const float* __restrict__ dinv,
                           const float* __restrict__ bias,
                           float* __restrict__ out) {
    int idx = blockIdx.x * blockDim.x + threadIdx.x;
    if (idx < N_NODES * DIM) {
        int i = idx >> 7;
        int cI = idx & (DIM - 1);
        float di = dinv[i];
        out[idx] = h[idx] * di * di + bias[cI];
    }
}

// ---------------- edge scatter: out[dst] += h[src] * norm ----------------
// One wave per edge: 32 lanes x float4 = 128 features, 512B coalesced gather.

__global__ __launch_bounds__(256) void k_scatter(const float* __restrict__ h,
                                                 const float* __restrict__ dinv,
                                                 const int* __restrict__ ei,
                                                 float* __restrict__ out) {
    int e    = (blockIdx.x * blockDim.x + threadIdx.x) >> 5;
    int lane = threadIdx.x & 31;
    if (e >= N_EDGES) return;

    int s = ei[e];
    int d = ei[N_EDGES + e];
    float norm = dinv[s] * dinv[d];

    const float4 m = *((const float4*)(h + (size_t)s * DIM) + lane);
    float* op = out + (size_t)d * DIM + lane * 4;
    unsafeAtomicAdd(op + 0, m.x * norm);
    unsafeAtomicAdd(op + 1, m.y * norm);
    unsafeAtomicAdd(op + 2, m.z * norm);
    unsafeAtomicAdd(op + 3, m.w * norm);
}

__global__ void k_relu(float* __restrict__ out) {
    int idx = blockIdx.x * blockDim.x + threadIdx.x;
    if (idx < N_NODES * DIM) out[idx] = fmaxf(out[idx], 0.0f);
}

// ---------------- launch ----------------

extern "C" void kernel_launch(void* const* d_in, const int* in_sizes, int n_in,
                              void* d_out, int out_size, void* d_ws, size_t ws_size,
                              hipStream_t stream) {
    const float* x    = (const float*)d_in[0];   // [N, 128]
    const float* W    = (const float*)d_in[1];   // [128, 128]
    const float* bias = (const float*)d_in[2];   // [128]
    const int*   ei   = (const int*)d_in[3];     // [2, E] flat
    float* out = (float*)d_out;                  // [N, 128]

    float* h   = (float*)d_ws;                   // N*128 floats = 25.6 MB
    float* deg = h + (size_t)N_NODES * DIM;      // N floats (deg -> dinv in place)

    const int T = 256;
    k_init_deg <<<(N_NODES + T - 1) / T, T, 0, stream>>>(deg);
    k_count_deg<<<(N_EDGES + T - 1) / T, T, 0, stream>>>(ei, deg);
    k_dinv     <<<(N_NODES + T - 1) / T, T, 0, stream>>>(deg);
    k_gemm     <<<N_NODES / 16,           T, 0, stream>>>(x, W, h);
    k_init_out <<<(N_NODES * DIM + T - 1) / T, T, 0, stream>>>(h, deg, bias, out);
    k_scatter  <<<(N_EDGES * 32 + T - 1) / T,  T, 0, stream>>>(h, deg, ei, out);
    k_relu     <<<(N_NODES * DIM + T - 1) / T, T, 0, stream>>>(out);
}